// GCN_21964462752266
// MI455X (gfx1250) — compile-verified
//
#include <hip/hip_runtime.h>
#include <hip/hip_bf16.h>

typedef __attribute__((ext_vector_type(2))) float v2f;
typedef __attribute__((ext_vector_type(8))) float v8f;

#define IN_F  128
#define H_F   128
#define OUT_F 64
#define EPS   1e-5f
// padded LDS row stride: 132 % 64 = 4 -> WMMA column reads hit distinct banks,
// and 132*4 = 528 bytes keeps every row 16-byte aligned for float4 staging.
#define A_PAD 132

// ---------------------------------------------------------------- utilities
__global__ void zero_f32(float* __restrict__ p, long long n) {
    long long i = blockIdx.x * (long long)blockDim.x + threadIdx.x;
    if (i < n) p[i] = 0.0f;
}

// deg_out[src[e]] += 1 ; deg_in[dst[e]] += 1
__global__ void degree_kernel(const int* __restrict__ src, const int* __restrict__ dst,
                              float* __restrict__ degO, float* __restrict__ degI, int E) {
    int e = blockIdx.x * blockDim.x + threadIdx.x;
    if (e < E) {
        atomicAdd(&degO[src[e]], 1.0f);
        atomicAdd(&degI[dst[e]], 1.0f);
    }
}

// in-place deg -> rsqrt(max(deg,1))
__global__ void norm_kernel(float* __restrict__ nsrc, float* __restrict__ ndst, int N) {
    int i = blockIdx.x * blockDim.x + threadIdx.x;
    if (i < N) {
        nsrc[i] = rsqrtf(fmaxf(nsrc[i], 1.0f));
        ndst[i] = rsqrtf(fmaxf(ndst[i], 1.0f));
    }
}

// ------------------------------------------------- GEMM1: h = (x*nsrc) @ W1
// block = 256 threads = 8 waves; block computes 16 rows x 128 cols.
// wave w computes the 16x16 tile at columns [16w,16w+16). K=128 in steps of 4.
__global__ __launch_bounds__(256) void gemm1_wmma(
    const float* __restrict__ x, const float* __restrict__ nsrc,
    const float* __restrict__ W1, float* __restrict__ h, int N) {
    __shared__ float As[16 * A_PAD];

    const int rowBase = blockIdx.x * 16;
    const int tid = threadIdx.x;
    const bool fullTile = (rowBase + 16 <= N);

    // stage A tile (scaled by norm_src) into LDS via float4: 512 vec4 / 256 thr
    if (fullTile) {
#pragma unroll
        for (int i = tid; i < 16 * (IN_F / 4); i += 256) {
            int r = i >> 5, c4 = i & 31;
            int gr = rowBase + r;
            float4 v = ((const float4*)x)[(long long)gr * (IN_F / 4) + c4];
            float s = nsrc[gr];
            v.x *= s; v.y *= s; v.z *= s; v.w *= s;
            *(float4*)&As[r * A_PAD + c4 * 4] = v;
        }
    } else {
        for (int i = tid; i < 16 * IN_F; i += 256) {
            int r = i >> 7, c = i & 127;
            int gr = rowBase + r;
            float v = 0.0f;
            if (gr < N) v = x[(long long)gr * IN_F + c] * nsrc[gr];
            As[r * A_PAD + c] = v;
        }
    }
    __syncthreads();

    const int wave = tid >> 5;
    const int lane = tid & 31;
    const int colBase = wave * 16;
    const int m    = lane & 15;        // A row
    const int n    = lane & 15;        // B/C column
    const int kOff = (lane >> 4) * 2;  // 0 or 2

    v8f acc = {};
#pragma unroll 4
    for (int k0 = 0; k0 < IN_F; k0 += 4) {
        v2f a, b;
        a.x = As[m * A_PAD + k0 + kOff];
        a.y = As[m * A_PAD + k0 + kOff + 1];
        b.x = W1[(k0 + kOff)     * H_F + colBase + n];
        b.y = W1[(k0 + kOff + 1) * H_F + colBase + n];
        acc = __builtin_amdgcn_wmma_f32_16x16x4_f32(false, a, false, b,
                                                    (short)0, acc, false, false);
    }

    // C/D layout: vgpr r holds row r (lanes 0-15) / row 8+r (lanes 16-31)
    const int mHi = (lane >> 4) * 8;
    float* hp = h + (long long)(rowBase + mHi) * H_F + colBase + n;
    if (fullTile) {
#pragma unroll
        for (int r = 0; r < 8; ++r) hp[r * H_F] = acc[r];
    } else {
#pragma unroll
        for (int r = 0; r < 8; ++r)
            if (rowBase + mHi + r < N) hp[r * H_F] = acc[r];
    }
}

// ------------------------------------- edge scatter: agg[dst] += h[src], F feats
// LANES lanes per edge, float4 gather + 4 scalar f32 atomics per lane
template <int F, int LANES>
__global__ void scatter_kernel(const float* __restrict__ h, const int* __restrict__ src,
                               const int* __restrict__ dst, float* __restrict__ agg, int E) {
    long long idx = blockIdx.x * (long long)blockDim.x + threadIdx.x;
    int e = (int)(idx / LANES);
    int t = (int)(idx % LANES);
    if (e >= E) return;
    int s = src[e], d = dst[e];
    float4 v = ((const float4*)(h + (long long)s * F))[t];
    float* ap = agg + (long long)d * F + t * 4;
    atomicAdd(ap + 0, v.x);
    atomicAdd(ap + 1, v.y);
    atomicAdd(ap + 2, v.z);
    atomicAdd(ap + 3, v.w);
}

// -------------------- conv1 epilogue + BN partial stats (block-reduced atomics)
// h1[n,f] = agg1[n,f]*ndst[n] + b1[f] (in place over agg1); accumulate sum/sumsq.
__global__ __launch_bounds__(128) void bn_stats_kernel(
    float* __restrict__ agg1, const float* __restrict__ ndst,
    const float* __restrict__ b1, float* __restrict__ sum,
    float* __restrict__ sumsq, int N) {
    int f = threadIdx.x;               // 0..127
    float bias = b1[f];
    float s = 0.0f, sq = 0.0f;
    for (int nrow = blockIdx.x; nrow < N; nrow += gridDim.x) {
        float v = agg1[(long long)nrow * H_F + f] * ndst[nrow] + bias;
        agg1[(long long)nrow * H_F + f] = v;
        s += v;
        sq += v * v;
    }
    atomicAdd(&sum[f], s);
    atomicAdd(&sumsq[f], sq);
}

// fold mean/var/gamma/beta into per-feature scale+shift
__global__ void bn_finalize_kernel(const float* __restrict__ sum, const float* __restrict__ sumsq,
                                   const float* __restrict__ gamma, const float* __restrict__ beta,
                                   float* __restrict__ bnsc, float* __restrict__ bnsh, int N) {
    int f = threadIdx.x;
    float invN = 1.0f / (float)N;
    float mean = sum[f] * invN;
    float var  = sumsq[f] * invN - mean * mean;
    float sc   = gamma[f] * rsqrtf(var + EPS);
    bnsc[f] = sc;
    bnsh[f] = beta[f] - mean * sc;
}

// --------- GEMM2: t2 = relu(bn(h1)) * nsrc @ W2  (BN+ReLU fused into A staging)
// block = 128 threads = 4 waves; 16 rows x 64 cols per block.
__global__ __launch_bounds__(128) void gemm2_wmma(
    const float* __restrict__ h1, const float* __restrict__ nsrc,
    const float* __restrict__ bnsc, const float* __restrict__ bnsh,
    const float* __restrict__ W2, float* __restrict__ t2, int N) {
    __shared__ float As[16 * A_PAD];

    const int rowBase = blockIdx.x * 16;
    const int tid = threadIdx.x;
    const bool fullTile = (rowBase + 16 <= N);

    if (fullTile) {
#pragma unroll
        for (int i = tid; i < 16 * (H_F / 4); i += 128) {
            int r = i >> 5, c4 = i & 31;
            int gr = rowBase + r;
            float4 v  = ((const float4*)h1)[(long long)gr * (H_F / 4) + c4];
            float4 sc = ((const float4*)bnsc)[c4];
            float4 sh = ((const float4*)bnsh)[c4];
            float ns  = nsrc[gr];
            v.x = fmaxf(v.x * sc.x + sh.x, 0.0f) * ns;
            v.y = fmaxf(v.y * sc.y + sh.y, 0.0f) * ns;
            v.z = fmaxf(v.z * sc.z + sh.z, 0.0f) * ns;
            v.w = fmaxf(v.w * sc.w + sh.w, 0.0f) * ns;
            *(float4*)&As[r * A_PAD + c4 * 4] = v;
        }
    } else {
        for (int i = tid; i < 16 * H_F; i += 128) {
            int r = i >> 7, c = i & 127;
            int gr = rowBase + r;
            float v = 0.0f;
            if (gr < N) {
                v = h1[(long long)gr * H_F + c] * bnsc[c] + bnsh[c];
                v = fmaxf(v, 0.0f) * nsrc[gr];
            }
            As[r * A_PAD + c] = v;
        }
    }
    __syncthreads();

    const int wave = tid >> 5;         // 0..3 -> col tile of OUT=64
    const int lane = tid & 31;
    const int colBase = wave * 16;
    const int m    = lane & 15;
    const int n    = lane & 15;
    const int kOff = (lane >> 4) * 2;

    v8f acc = {};
#pragma unroll 4
    for (int k0 = 0; k0 < H_F; k0 += 4) {
        v2f a, b;
        a.x = As[m * A_PAD + k0 + kOff];
        a.y = As[m * A_PAD + k0 + kOff + 1];
        b.x = W2[(k0 + kOff)     * OUT_F + colBase + n];
        b.y = W2[(k0 + kOff + 1) * OUT_F + colBase + n];
        acc = __builtin_amdgcn_wmma_f32_16x16x4_f32(false, a, false, b,
                                                    (short)0, acc, false, false);
    }

    const int mHi = (lane >> 4) * 8;
    float* tp = t2 + (long long)(rowBase + mHi) * OUT_F + colBase + n;
    if (fullTile) {
#pragma unroll
        for (int r = 0; r < 8; ++r) tp[r * OUT_F] = acc[r];
    } else {
#pragma unroll
        for (int r = 0; r < 8; ++r)
            if (rowBase + mHi + r < N) tp[r * OUT_F] = acc[r];
    }
}

// out[n,f] = agg2[n,f]*ndst[n] + b2[f]
__global__ void final_kernel(const float* __restrict__ agg2, const float* __restrict__ ndst,
                             const float* __restrict__ b2, float* __restrict__ out, int N) {
    long long i = blockIdx.x * (long long)blockDim.x + threadIdx.x;
    if (i < (long long)N * OUT_F) {
        int nrow = (int)(i >> 6);
        int f = (int)(i & 63);
        out[i] = agg2[i] * ndst[nrow] + b2[f];
    }
}

// ---------------------------------------------------------------- launcher
extern "C" void kernel_launch(void* const* d_in, const int* in_sizes, int n_in,
                              void* d_out, int out_size, void* d_ws, size_t ws_size,
                              hipStream_t stream) {
    const float* x     = (const float*)d_in[0];
    const int*   src   = (const int*)d_in[1];
    const int*   dst   = (const int*)d_in[2];
    const float* W1    = (const float*)d_in[3];
    const float* b1    = (const float*)d_in[4];
    const float* gamma = (const float*)d_in[5];
    const float* beta  = (const float*)d_in[6];
    const float* W2    = (const float*)d_in[7];
    const float* b2    = (const float*)d_in[8];
    float*       out   = (float*)d_out;

    const int N = in_sizes[0] / IN_F;
    const int E = in_sizes[1];

    // workspace layout (floats)
    float* ws    = (float*)d_ws;
    float* nsrc  = ws;                              // N   (deg_out -> norm_src in place)
    float* ndst  = nsrc + N;                        // N   (deg_in  -> norm_dst in place)
    float* h     = ndst + N;                        // N*128
    float* agg1  = h    + (size_t)N * H_F;          // N*128 (becomes h1 in place)
    float* t2    = agg1 + (size_t)N * H_F;          // N*64
    float* agg2  = t2   + (size_t)N * OUT_F;        // N*64
    float* sum   = agg2 + (size_t)N * OUT_F;        // 128
    float* sumsq = sum + 128;                       // 128
    float* bnsc  = sumsq + 128;                     // 128
    float* bnsh  = bnsc + 128;                      // 128

    const int TB = 256;
    // zero accumulators (harness does not re-poison between replays)
    {
        long long n0 = 2LL * N;
        zero_f32<<<(int)((n0 + TB - 1) / TB), TB, 0, stream>>>(nsrc, n0);
        long long n1 = (long long)N * H_F;
        zero_f32<<<(int)((n1 + TB - 1) / TB), TB, 0, stream>>>(agg1, n1);
        long long n2 = (long long)N * OUT_F;
        zero_f32<<<(int)((n2 + TB - 1) / TB), TB, 0, stream>>>(agg2, n2);
        zero_f32<<<1, 512, 0, stream>>>(sum, 512);
    }

    // degrees and norms
    degree_kernel<<<(E + TB - 1) / TB, TB, 0, stream>>>(src, dst, nsrc, ndst, E);
    norm_kernel<<<(N + TB - 1) / TB, TB, 0, stream>>>(nsrc, ndst, N);

    // conv1 dense part (WMMA)
    const int rowTiles = (N + 15) / 16;
    gemm1_wmma<<<rowTiles, 256, 0, stream>>>(x, nsrc, W1, h, N);

    // conv1 edge aggregation: 32 lanes/edge, float4 gather + f32 atomics
    {
        long long th = (long long)E * 32;
        scatter_kernel<H_F, 32><<<(int)((th + TB - 1) / TB), TB, 0, stream>>>(h, src, dst, agg1, E);
    }

    // conv1 epilogue + BN stats, then fold BN params
    bn_stats_kernel<<<1024, 128, 0, stream>>>(agg1, ndst, b1, sum, sumsq, N);
    bn_finalize_kernel<<<1, 128, 0, stream>>>(sum, sumsq, gamma, beta, bnsc, bnsh, N);

    // conv2 dense part (WMMA) with fused BN+ReLU+norm_src on A
    gemm2_wmma<<<rowTiles, 128, 0, stream>>>(agg1, nsrc, bnsc, bnsh, W2, t2, N);

    // conv2 edge aggregation: 16 lanes/edge
    {
        long long th = (long long)E * 16;
        scatter_kernel<OUT_F, 16><<<(int)((th + TB - 1) / TB), TB, 0, stream>>>(t2, src, dst, agg2, E);
    }

    // final scale + bias
    {
        long long th = (long long)N * OUT_F;
        final_kernel<<<(int)((th + TB - 1) / TB), TB, 0, stream>>>(agg2, ndst, b2, out, N);
    }
}